// TFModel_54434415510065
// MI455X (gfx1250) — compile-verified
//
#include <hip/hip_runtime.h>
#include <cstdint>

typedef __attribute__((ext_vector_type(2))) float v2f;
typedef __attribute__((ext_vector_type(8))) float v8f;

#define HW2   1156          // 34*34
#define NB1   17340         // 15*34*34
#define IMC   269.0f        // IMW-1 = IMH-1
#define KP    512           // padded K pitch for GEMM1 (490 -> 512, zero-filled)
#define FEATN 2048

__device__ __constant__ float c_AW[15] = {9.232984f,16.0f,27.712813f,18.465969f,32.0f,55.425626f,36.931937f,64.0f,110.851252f,73.863875f,128.0f,221.702503f,147.72775f,256.0f,443.405007f};
__device__ __constant__ float c_AH[15] = {27.72668f,16.0f,9.237604f,55.453359f,32.0f,18.475209f,110.906719f,64.0f,36.950417f,221.813438f,128.0f,73.900834f,443.626876f,256.0f,147.801669f};
__device__ __constant__ float c_RPN_MEAN[4] = {0.000437f,0.002586f,-0.123953f,-0.081469f};
__device__ __constant__ float c_RPN_STD[4]  = {0.12677f,0.095741f,0.3173f,0.281042f};

__device__ __forceinline__ float clampf(float v, float lo, float hi){ return fminf(fmaxf(v,lo),hi); }
// monotonic float -> uint key (ascending)
__device__ __forceinline__ uint32_t fkey(float s){
  uint32_t u = __float_as_uint(s);
  return (u & 0x80000000u) ? ~u : (u | 0x80000000u);
}

// ---------------- Stage 1: RPN decode + keep-mask ----------------
__global__ void k_rpn_decode(const float* __restrict__ cls, const float* __restrict__ bb,
                             float* __restrict__ boxes, float* __restrict__ scores,
                             uint32_t* __restrict__ keys)
{
  int i = blockIdx.x*blockDim.x + threadIdx.x;
  if (i >= NB1) return;
  int a = i / HW2, rem = i % HW2;
  int y = rem / 34, x = rem % 34;
  float fg = cls[(15 + a)*HW2 + rem];
  float t0 = bb[(a*4+0)*HW2 + rem]*c_RPN_STD[0] + c_RPN_MEAN[0];
  float t1 = bb[(a*4+1)*HW2 + rem]*c_RPN_STD[1] + c_RPN_MEAN[1];
  float t2 = bb[(a*4+2)*HW2 + rem]*c_RPN_STD[2] + c_RPN_MEAN[2];
  float t3 = bb[(a*4+3)*HW2 + rem]*c_RPN_STD[3] + c_RPN_MEAN[3];
  float aw = c_AW[a], ah = c_AH[a];
  float cx = t0*aw + (float)x*8.0f;
  float cy = t1*ah + (float)y*8.0f;
  float bw = expf(t2)*aw, bh = expf(t3)*ah;
  float x1 = clampf(cx - 0.5f*(bw-1.0f), 0.f, IMC);
  float y1 = clampf(cy - 0.5f*(bh-1.0f), 0.f, IMC);
  float x2 = clampf(cx + 0.5f*(bw-1.0f), 0.f, IMC);
  float y2 = clampf(cy + 0.5f*(bh-1.0f), 0.f, IMC);
  boxes[i*4+0]=x1; boxes[i*4+1]=y1; boxes[i*4+2]=x2; boxes[i*4+3]=y2;
  float w = x2-x1+1.0f, h = y2-y1+1.0f;
  bool keep = (fg >= 0.2f) && ((w >= 6.16056f) || (h >= 6.16056f));
  float s = keep ? fg : -1.0f;
  scores[i] = s;
  keys[i] = fkey(s);
}

// ---------------- Stage 2: radix-select top-3000 (single block) ----------------
__global__ void k_select3000(const uint32_t* __restrict__ keys, const float* __restrict__ scores,
                             const float* __restrict__ boxes,
                             float* __restrict__ candBox, float* __restrict__ candScore)
{
  __shared__ uint32_t hist[256];
  __shared__ uint32_t s_prefix, s_want;
  __shared__ uint32_t cntG, cntE;
  const int t = threadIdx.x, NT = blockDim.x;
  if (t == 0){ s_prefix = 0u; s_want = 3000u; cntG = 0u; cntE = 0u; }
  __syncthreads();
  uint32_t mask = 0u;
  for (int r = 0; r < 4; ++r){
    int shift = 24 - 8*r;
    if (t < 256) hist[t] = 0u;
    __syncthreads();
    uint32_t pref = s_prefix;
    for (int i = t; i < NB1; i += NT){
      uint32_t k = keys[i];
      if ((k & mask) == pref) atomicAdd(&hist[(k >> shift) & 0xFFu], 1u);
    }
    __syncthreads();
    if (t == 0){
      uint32_t cum = 0u, want = s_want, sel = 0u;
      for (int b = 255; b >= 0; --b){
        uint32_t h = hist[b];
        if (cum + h >= want){ sel = (uint32_t)b; break; }
        cum += h;
      }
      s_want = want - cum;
      s_prefix = pref | (sel << shift);
    }
    __syncthreads();
    mask |= (0xFFu << shift);
  }
  uint32_t T = s_prefix;   // 3000th-largest key
  __syncthreads();
  // all strictly greater keys are in (guaranteed < 3000 of them)
  for (int i = t; i < NB1; i += NT){
    if (keys[i] > T){
      uint32_t pos = atomicAdd(&cntG, 1u);
      candBox[pos*4+0]=boxes[i*4+0]; candBox[pos*4+1]=boxes[i*4+1];
      candBox[pos*4+2]=boxes[i*4+2]; candBox[pos*4+3]=boxes[i*4+3];
      candScore[pos]=scores[i];
    }
  }
  __syncthreads();
  uint32_t base = cntG;
  for (int i = t; i < NB1; i += NT){
    if (keys[i] == T){
      uint32_t pos = base + atomicAdd(&cntE, 1u);
      if (pos < 3000u){
        candBox[pos*4+0]=boxes[i*4+0]; candBox[pos*4+1]=boxes[i*4+1];
        candBox[pos*4+2]=boxes[i*4+2]; candBox[pos*4+3]=boxes[i*4+3];
        candScore[pos]=scores[i];
      }
    }
  }
}

// ---------------- Stage 3: greedy NMS 0.7 -> 300 rois (single block) ----------------
__global__ void k_nms1(const float* __restrict__ candBox, const float* __restrict__ candScore,
                       float* __restrict__ rois_xy, float* __restrict__ validF,
                       float* __restrict__ out)
{
  __shared__ float sc[3000];
  __shared__ float rv[1024];
  __shared__ int   ri[1024];
  __shared__ float bj[4];
  __shared__ int   selIdx[300];
  __shared__ float selValid[300];
  const int t = threadIdx.x, NT = blockDim.x;   // 1024
  for (int i = t; i < 3000; i += NT) sc[i] = candScore[i];
  __syncthreads();
  for (int it = 0; it < 300; ++it){
    float bv = -1e30f; int bi = 0x7fffffff;
    for (int i = t; i < 3000; i += NT){ float v = sc[i]; if (v > bv){ bv = v; bi = i; } }
    rv[t] = bv; ri[t] = bi;
    __syncthreads();
    for (int s = NT >> 1; s > 0; s >>= 1){
      if (t < s){
        float ov = rv[t+s]; int oi = ri[t+s];
        if (ov > rv[t] || (ov == rv[t] && oi < ri[t])){ rv[t] = ov; ri[t] = oi; }
      }
      __syncthreads();
    }
    int j = ri[0]; float vj = rv[0];
    if (t == 0){
      selIdx[it] = j; selValid[it] = (vj > 0.f) ? 1.f : 0.f;
      bj[0]=candBox[j*4+0]; bj[1]=candBox[j*4+1]; bj[2]=candBox[j*4+2]; bj[3]=candBox[j*4+3];
    }
    __syncthreads();
    float jx1=bj[0], jy1=bj[1], jx2=bj[2], jy2=bj[3];
    float areaJ = (jx2-jx1+1.f)*(jy2-jy1+1.f);
    for (int i = t; i < 3000; i += NT){
      float x1=candBox[i*4+0], y1=candBox[i*4+1], x2=candBox[i*4+2], y2=candBox[i*4+3];
      float area = (x2-x1+1.f)*(y2-y1+1.f);
      float iw = fmaxf(fminf(x2,jx2) - fmaxf(x1,jx1) + 1.f, 0.f);
      float ih = fmaxf(fminf(y2,jy2) - fmaxf(y1,jy1) + 1.f, 0.f);
      float inter = iw*ih;
      float iou = inter / (area + areaJ - inter);
      if (iou > 0.7f || i == j) sc[i] = -1.f;
    }
    __syncthreads();
  }
  for (int r = t; r < 300; r += NT){
    int j = selIdx[r]; float vm = selValid[r];
    float x1=candBox[j*4+0]*vm, y1=candBox[j*4+1]*vm, x2=candBox[j*4+2]*vm, y2=candBox[j*4+3]*vm;
    rois_xy[r*4+0]=x1; rois_xy[r*4+1]=y1; rois_xy[r*4+2]=x2; rois_xy[r*4+3]=y2;
    validF[r] = vm;
    out[30 + r*5 + 0] = 0.f;
    out[30 + r*5 + 1] = x1; out[30 + r*5 + 2] = y1;
    out[30 + r*5 + 3] = x2; out[30 + r*5 + 4] = y2;
    out[1530 + r] = candScore[j]*vm;
  }
}

// ---------------- Stage 4: PSROI pooling (one block per ROI) ----------------
__global__ void k_psroi(const float* __restrict__ ft, const float* __restrict__ rois_xy,
                        float* __restrict__ pooledP)
{
  int r = blockIdx.x;
  int t = threadIdx.x;
  if (t >= 490) return;            // t == c*49 + p*7 + q
  int rem = t % 49, p = rem / 7, q = rem % 7;
  const float* roi = rois_xy + r*4;
  float sw = roi[0]*0.125f, sh = roi[1]*0.125f;
  float ew = (roi[2]+1.0f)*0.125f, eh = (roi[3]+1.0f)*0.125f;
  float rh = fmaxf(eh - sh, 0.1f), rw = fmaxf(ew - sw, 0.1f);
  float bh = rh*(1.0f/7.0f), bw = rw*(1.0f/7.0f);
  const float* plane = ft + t*HW2;
  float sum = 0.f;
  #pragma unroll
  for (int sy = 0; sy < 4; ++sy){
    float yy = sh + p*bh + (sy+0.5f)*(bh*0.25f);
    float y0 = floorf(yy); float dy = yy - y0;
    int y0i = (int)y0;
    float wy0 = 1.f - dy, wy1 = dy;
    float my0 = (y0i   >= 0 && y0i   < 34) ? 1.f : 0.f;
    float my1 = (y0i+1 >= 0 && y0i+1 < 34) ? 1.f : 0.f;
    int yc0 = min(max(y0i,   0), 33);
    int yc1 = min(max(y0i+1, 0), 33);
    #pragma unroll
    for (int sx = 0; sx < 4; ++sx){
      float xx = sw + q*bw + (sx+0.5f)*(bw*0.25f);
      float x0 = floorf(xx); float dx = xx - x0;
      int x0i = (int)x0;
      float wx0 = 1.f - dx, wx1 = dx;
      float mx0 = (x0i   >= 0 && x0i   < 34) ? 1.f : 0.f;
      float mx1 = (x0i+1 >= 0 && x0i+1 < 34) ? 1.f : 0.f;
      int xc0 = min(max(x0i,   0), 33);
      int xc1 = min(max(x0i+1, 0), 33);
      sum += plane[yc0*34+xc0]*(wy0*my0)*(wx0*mx0)
           + plane[yc0*34+xc1]*(wy0*my0)*(wx1*mx1)
           + plane[yc1*34+xc0]*(wy1*my1)*(wx0*mx0)
           + plane[yc1*34+xc1]*(wy1*my1)*(wx1*mx1);
    }
  }
  pooledP[r*KP + t] = sum * (1.0f/16.0f);
}

// ---------------- W_inner staging into zero-padded WinP[2048][512] ----------------
__global__ void k_prep_win(const float* __restrict__ W, float* __restrict__ WinP)
{
  int idx = blockIdx.x*blockDim.x + threadIdx.x;   // exactly 2048*512 threads
  int n = idx >> 9;          // /512
  int k = idx & 511;
  WinP[idx] = (k < 490) ? W[n*490 + k] : 0.f;
}

// ---------------- combined cls+bbox weight staging: Wcb[32][2048], bcb[32] ----------------
__global__ void k_prep_wcb(const float* __restrict__ Wcls, const float* __restrict__ Wbbox,
                           const float* __restrict__ bcls, const float* __restrict__ bbbox,
                           float* __restrict__ Wcb, float* __restrict__ bcb)
{
  int idx = blockIdx.x*blockDim.x + threadIdx.x;   // exactly 32*2048 threads
  int n = idx >> 11;
  int k = idx & 2047;
  float v = 0.f;
  if (n < 4)       v = Wcls[n*2048 + k];
  else if (n < 20) v = Wbbox[(n-4)*2048 + k];
  Wcb[idx] = v;
  if (idx < 32) bcb[idx] = (idx < 4) ? bcls[idx] : ((idx < 20) ? bbbox[idx-4] : 0.f);
}

// ---------------- Stage 5: FC GEMM 304x2048x512 via V_WMMA_F32_16X16X4_F32 + ReLU ----------------
// Both operands zero-padded in K: no guards, no exec divergence in the hot loop.
__global__ void k_gemm1(const float* __restrict__ pooledP, const float* __restrict__ WinP,
                        const float* __restrict__ bias, float* __restrict__ featP)
{
  const int lane = threadIdx.x & 31;
  const int wv   = threadIdx.x >> 5;
  const int tile = blockIdx.x * 8 + wv;   // 0..2431 exactly (304 blocks * 8 waves)
  const int tM = tile >> 7;               // 0..18
  const int tN = tile & 127;              // 0..127
  const int Mb = tM*16, Nb = tN*16;
  const int half = lane >> 4, l15 = lane & 15;
  const int n = Nb + l15;
  const float* Ar = pooledP + (Mb + l15)*KP + half*2;  // A: lane row, K pairs per half-wave
  const float* Br = WinP    + n*KP          + half*2;  // B: lane col (n), K pairs per half-wave
  const float bn = bias[n];
  v8f acc;
  #pragma unroll
  for (int v = 0; v < 8; ++v) acc[v] = bn;    // C = broadcast bias along M
  #pragma unroll 4
  for (int k = 0; k < KP; k += 4){
    v2f a, b;
    a.x = Ar[k]; a.y = Ar[k+1];
    b.x = Br[k]; b.y = Br[k+1];
    acc = __builtin_amdgcn_wmma_f32_16x16x4_f32(false, a, false, b, (short)0, acc, false, false);
  }
  #pragma unroll
  for (int v = 0; v < 8; ++v){
    const int m = Mb + half*8 + v;          // C/D layout: VGPR v, halves offset by 8 rows
    const float x = acc[v];
    featP[m*FEATN + n] = x > 0.f ? x : 0.f;
  }
}

// ---------------- Stage 6: head GEMM 304x32x2048 (cls 0..3, bbox 4..19) ----------------
__global__ void k_gemm2(const float* __restrict__ featP, const float* __restrict__ Wcb,
                        const float* __restrict__ bcb, float* __restrict__ logits)
{
  const int lane = threadIdx.x & 31;
  const int wv   = threadIdx.x >> 5;
  const int tile = blockIdx.x * 8 + wv;
  if (tile >= 38) return;                 // wave-uniform exit, EXEC stays all-ones for WMMA
  const int tM = tile >> 1, tN = tile & 1;
  const int Mb = tM*16, Nb = tN*16;
  const int half = lane >> 4, l15 = lane & 15;
  const int n = Nb + l15;
  const float* Ar = featP + (Mb + l15)*FEATN + half*2;
  const float* Br = Wcb   + n*FEATN          + half*2;
  const float bn = bcb[n];
  v8f acc;
  #pragma unroll
  for (int v = 0; v < 8; ++v) acc[v] = bn;
  #pragma unroll 4
  for (int k = 0; k < FEATN; k += 4){
    v2f a, b;
    a.x = Ar[k]; a.y = Ar[k+1];
    b.x = Br[k]; b.y = Br[k+1];
    acc = __builtin_amdgcn_wmma_f32_16x16x4_f32(false, a, false, b, (short)0, acc, false, false);
  }
  #pragma unroll
  for (int v = 0; v < 8; ++v){
    const int m = Mb + half*8 + v;
    logits[m*32 + n] = acc[v];
  }
}

// ---------------- Stage 7: softmax + class decode + keep mask (900 candidates) ----------------
__global__ void k_stage2(const float* __restrict__ logits, const float* __restrict__ rois_xy,
                         const float* __restrict__ validF,
                         float* __restrict__ cb, float* __restrict__ cl, float* __restrict__ ms)
{
  int idx = blockIdx.x*blockDim.x + threadIdx.x;
  if (idx >= 900) return;
  int c = idx/300 + 1;         // class 1..3 (class-major concatenation)
  int r = idx%300;
  const float* L = logits + r*32;
  float mx = fmaxf(fmaxf(L[0],L[1]), fmaxf(L[2],L[3]));
  float e0=expf(L[0]-mx), e1=expf(L[1]-mx), e2=expf(L[2]-mx), e3=expf(L[3]-mx);
  float inv = 1.0f/(e0+e1+e2+e3);
  float p0 = e0*inv;
  float pc = ((c==1) ? e1 : (c==2) ? e2 : e3) * inv;
  float rx1=rois_xy[r*4+0], ry1=rois_xy[r*4+1], rx2=rois_xy[r*4+2], ry2=rois_xy[r*4+3];
  float rw  = rx2-rx1+1.f, rh2 = ry2-ry1+1.f;
  float rcx = rx1 + 0.5f*(rw-1.f), rcy = ry1 + 0.5f*(rh2-1.f);
  // bpred col 4c+ch == combined logits col 4 + 4c + ch ; RCNN_STD = (.1,.1,.2,.2)
  float t0 = L[4 + 4*c + 0]*0.1f;
  float t1 = L[4 + 4*c + 1]*0.1f;
  float t2 = L[4 + 4*c + 2]*0.2f;
  float t3 = L[4 + 4*c + 3]*0.2f;
  float cx = t0*rw + rcx, cy = t1*rh2 + rcy;
  float bw = expf(t2)*rw, bh = expf(t3)*rh2;
  float x1 = clampf(cx-0.5f*(bw-1.f), 0.f, IMC);
  float y1 = clampf(cy-0.5f*(bh-1.f), 0.f, IMC);
  float x2 = clampf(cx+0.5f*(bw-1.f), 0.f, IMC);
  float y2 = clampf(cy+0.5f*(bh-1.f), 0.f, IMC);
  float dw = x2-x1+1.f, dh = y2-y1+1.f;
  bool keep = (validF[r] > 0.5f) && ((1.f - p0) >= 0.1f) && (pc >= 0.1f)
              && ((dw >= 8.8008f) || (dh >= 8.8008f));
  cb[idx*4+0]=x1; cb[idx*4+1]=y1; cb[idx*4+2]=x2; cb[idx*4+3]=y2;
  cl[idx] = (float)c;
  ms[idx] = keep ? pc : -1.0f;
}

// ---------------- Stage 8: NMS 0.5 -> top 5 final boxes (single block) ----------------
__global__ void k_nms2(const float* __restrict__ cb, const float* __restrict__ cl,
                       const float* __restrict__ ms, float* __restrict__ out)
{
  __shared__ float sc[900];
  __shared__ float rv[256];
  __shared__ int   ri[256];
  __shared__ float bj[4];
  __shared__ int   selIdx[5];
  __shared__ float selValid[5];
  const int t = threadIdx.x, NT = 256;
  for (int i = t; i < 900; i += NT) sc[i] = ms[i];
  __syncthreads();
  for (int it = 0; it < 5; ++it){
    float bv = -1e30f; int bi = 0x7fffffff;
    for (int i = t; i < 900; i += NT){ float v = sc[i]; if (v > bv){ bv = v; bi = i; } }
    rv[t] = bv; ri[t] = bi;
    __syncthreads();
    for (int s = NT >> 1; s > 0; s >>= 1){
      if (t < s){
        float ov = rv[t+s]; int oi = ri[t+s];
        if (ov > rv[t] || (ov == rv[t] && oi < ri[t])){ rv[t] = ov; ri[t] = oi; }
      }
      __syncthreads();
    }
    int j = ri[0]; float vj = rv[0];
    if (t == 0){
      selIdx[it] = j; selValid[it] = (vj > 0.f) ? 1.f : 0.f;
      bj[0]=cb[j*4+0]; bj[1]=cb[j*4+1]; bj[2]=cb[j*4+2]; bj[3]=cb[j*4+3];
    }
    __syncthreads();
    float jx1=bj[0], jy1=bj[1], jx2=bj[2], jy2=bj[3];
    float areaJ = (jx2-jx1+1.f)*(jy2-jy1+1.f);
    for (int i = t; i < 900; i += NT){
      float x1=cb[i*4+0], y1=cb[i*4+1], x2=cb[i*4+2], y2=cb[i*4+3];
      float area = (x2-x1+1.f)*(y2-y1+1.f);
      float iw = fmaxf(fminf(x2,jx2) - fmaxf(x1,jx1) + 1.f, 0.f);
      float ih = fmaxf(fminf(y2,jy2) - fmaxf(y1,jy1) + 1.f, 0.f);
      float inter = iw*ih;
      float iou = inter / (area + areaJ - inter);
      if (iou > 0.5f || i == j) sc[i] = -1.f;
    }
    __syncthreads();
  }
  if (t < 5){
    int j = selIdx[t]; float v = selValid[t];
    out[t*6+0]=cb[j*4+0]*v; out[t*6+1]=cb[j*4+1]*v;
    out[t*6+2]=cb[j*4+2]*v; out[t*6+3]=cb[j*4+3]*v;
    out[t*6+4]=ms[j]*v;     out[t*6+5]=cl[j]*v;
  }
}

extern "C" void kernel_launch(void* const* d_in, const int* in_sizes, int n_in,
                              void* d_out, int out_size, void* d_ws, size_t ws_size,
                              hipStream_t stream)
{
  const float* rpn_cls = (const float*)d_in[0];   // (1,30,34,34)
  const float* rpn_bb  = (const float*)d_in[1];   // (1,60,34,34)
  const float* ft      = (const float*)d_in[2];   // (1,490,34,34)
  const float* Winner  = (const float*)d_in[3];   // (2048,490)
  const float* binner  = (const float*)d_in[4];   // (2048,)
  const float* Wcls    = (const float*)d_in[5];   // (4,2048)
  const float* bcls    = (const float*)d_in[6];   // (4,)
  const float* Wbbox   = (const float*)d_in[7];   // (16,2048)
  const float* bbbox   = (const float*)d_in[8];   // (16,)
  float* out = (float*)d_out;                     // 30 + 1500 + 300 = 1830 floats

  float* ws = (float*)d_ws;
  size_t off = 0;
  float*    boxes1    = ws + off; off += (size_t)NB1*4;
  float*    scores1   = ws + off; off += NB1;
  uint32_t* keys      = (uint32_t*)(ws + off); off += NB1;
  float*    candBox   = ws + off; off += 3000*4;
  float*    candScore = ws + off; off += 3000;
  float*    rois_xy   = ws + off; off += 300*4;
  float*    validF    = ws + off; off += 300;
  float*    pooledP   = ws + off; off += 304*KP;       // zero-padded A for GEMM1
  float*    WinP      = ws + off; off += (size_t)FEATN*KP;  // zero-padded B for GEMM1
  float*    featP     = ws + off; off += 304*FEATN;
  float*    Wcb       = ws + off; off += 32*FEATN;
  float*    bcb       = ws + off; off += 32;
  float*    logits    = ws + off; off += 304*32;
  float*    cb        = ws + off; off += 900*4;
  float*    cl        = ws + off; off += 900;
  float*    ms        = ws + off; off += 900;
  (void)ws_size; (void)in_sizes; (void)n_in; (void)out_size;

  hipMemsetAsync(pooledP, 0, (size_t)304*KP*sizeof(float), stream);
  k_rpn_decode<<<(NB1+255)/256, 256, 0, stream>>>(rpn_cls, rpn_bb, boxes1, scores1, keys);
  k_select3000<<<1, 1024, 0, stream>>>(keys, scores1, boxes1, candBox, candScore);
  k_nms1<<<1, 1024, 0, stream>>>(candBox, candScore, rois_xy, validF, out);
  k_psroi<<<300, 512, 0, stream>>>(ft, rois_xy, pooledP);
  k_prep_win<<<(FEATN*KP)/256, 256, 0, stream>>>(Winner, WinP);
  k_prep_wcb<<<256, 256, 0, stream>>>(Wcls, Wbbox, bcls, bbbox, Wcb, bcb);
  k_gemm1<<<304, 256, 0, stream>>>(pooledP, WinP, binner, featP);
  k_gemm2<<<5, 256, 0, stream>>>(featP, Wcb, bcb, logits);
  k_stage2<<<4, 256, 0, stream>>>(logits, rois_xy, validF, cb, cl, ms);
  k_nms2<<<1, 256, 0, stream>>>(cb, cl, ms, out);
}